// PrincipledLearningModel_11553462026866
// MI455X (gfx1250) — compile-verified
//
#include <hip/hip_runtime.h>
#include <hip/hip_bf16.h>
#include <math.h>

// ---------------------------------------------------------------------------
// Problem constants (match reference)
// ---------------------------------------------------------------------------
#define N_CLAUSES 50000
#define FEAT_DIM  64
#define EMBED_DIM 128
#define NUM_QUEUES 8
#define NUM_STEPS 2048
#define ENTROPY_COEF 0.1f

typedef __attribute__((ext_vector_type(2))) float f32x2;
typedef __attribute__((ext_vector_type(4))) float f32x4;
typedef __attribute__((ext_vector_type(8))) float f32x8;

// ---------------------------------------------------------------------------
// Kernel 0: fold keys @ W2^T  ->  KW[16][128]  (rows 8..15 zero-padded for the
// 16x16 WMMA), and kb[q] = keys[q] . b2.
//   emb[n][c] = sum_e h[n][e] * W2[e][c]
//   logits[q][n] = sum_c keys[q][c]*emb[n][c] = sum_e h[n][e]*KW[q][e] + kb[q]
//   with KW[q][e] = sum_c keys[q][c] * W2[e][c]
// ---------------------------------------------------------------------------
__global__ void plm_prep_kernel(const float* __restrict__ keys,
                                const float* __restrict__ W2,
                                const float* __restrict__ b2,
                                float* __restrict__ KW,
                                float* __restrict__ kb) {
  int q = blockIdx.x;  // 0..15
  for (int e = threadIdx.x; e < EMBED_DIM; e += blockDim.x) {
    float acc = 0.f;
    if (q < NUM_QUEUES) {
      for (int c = 0; c < EMBED_DIM; ++c)
        acc += keys[q * EMBED_DIM + c] * W2[e * EMBED_DIM + c];
    }
    KW[q * EMBED_DIM + e] = acc;
  }
  if (threadIdx.x == 0) {
    float acc = 0.f;
    if (q < NUM_QUEUES) {
      for (int c = 0; c < EMBED_DIM; ++c)
        acc += keys[q * EMBED_DIM + c] * b2[c];
    }
    kb[q] = acc;
  }
}

// ---------------------------------------------------------------------------
// Kernel 1: per 16-clause tile (one tile per wave32):
//   H = relu(X_tile @ W1 + b1)           via V_WMMA_F32_16X16X4_F32 (128 ops)
//   D = H @ KW^T                          via V_WMMA_F32_16X16X4_F32 (32 ops)
//   logits[q][n] = D[n][q] + kb[q]
// A-fragment (16x4 f32): lane l: M = l&15; VGPR j: K = k0 + j + 2*(l>>4)
// B/C/D: row striped across lanes; D element (vgpr v, lane l):
//   M = v + 8*(l>>4), N = l&15
// H is staged through per-wave LDS to convert D-layout -> A-layout (same-wave
// DS ops are in-order; no barrier needed).
// ---------------------------------------------------------------------------
#define WAVES_PER_BLOCK 4
#define N_TILES (N_CLAUSES / 16)  // 3125 exactly

__global__ void plm_embed_logits_kernel(const float* __restrict__ X,
                                        const float* __restrict__ W1,
                                        const float* __restrict__ b1,
                                        const float* __restrict__ KW,
                                        const float* __restrict__ kb,
                                        float* __restrict__ logits) {
  __shared__ float sKW[16 * EMBED_DIM];                   // 8 KB
  __shared__ float sH[WAVES_PER_BLOCK][16 * EMBED_DIM];   // 4 x 8 KB

  // Cooperative stage of KW into LDS.
  for (int i = threadIdx.x; i < 16 * EMBED_DIM; i += blockDim.x)
    sKW[i] = KW[i];
  __syncthreads();

  const int w    = threadIdx.x >> 5;         // wave within block
  const int l    = threadIdx.x & 31;         // lane
  const int ln   = l & 15;
  const int koff = (l >> 4) << 1;            // 0 or 2
  const int tile = blockIdx.x * WAVES_PER_BLOCK + w;
  if (tile >= N_TILES) return;               // no barriers below this point
  const int r0 = tile * 16;
  const int row = r0 + ln;                   // clause handled by this lane (A rows)

  // Preload this lane's A-fragment data for all 16 K-chunks of GEMM1.
  f32x2 areg[FEAT_DIM / 4];
#pragma unroll
  for (int kc = 0; kc < FEAT_DIM / 4; ++kc)
    areg[kc] = *(const f32x2*)(X + (size_t)row * FEAT_DIM + kc * 4 + koff);

  // ---- GEMM1: H = relu(X @ W1 + b1), written to per-wave LDS ----
  for (int nc = 0; nc < EMBED_DIM / 16; ++nc) {
    const int col = nc * 16 + ln;
    f32x8 acc = {};
#pragma unroll
    for (int kc = 0; kc < FEAT_DIM / 4; ++kc) {
      const int k0 = kc * 4;
      f32x2 b;
      b.x = W1[(k0 + koff)     * EMBED_DIM + col];
      b.y = W1[(k0 + koff + 1) * EMBED_DIM + col];
      acc = __builtin_amdgcn_wmma_f32_16x16x4_f32(
          /*neg_a=*/false, areg[kc], /*neg_b=*/false, b,
          /*c_mod=*/(short)0, acc, /*reuse_a=*/false, /*reuse_b=*/false);
    }
    const float bv = b1[col];
#pragma unroll
    for (int v = 0; v < 8; ++v) {
      const int m = v + ((l >> 4) << 3);     // clause within tile
      float h = acc[v] + bv;
      h = fmaxf(h, 0.0f);
      sH[w][m * EMBED_DIM + col] = h;        // relu'd hidden activation
    }
  }

  // ---- GEMM2: D = H(16x128) @ KW^T(128x16) ----
  f32x8 dacc = {};
#pragma unroll
  for (int kc = 0; kc < EMBED_DIM / 4; ++kc) {
    const int k0 = kc * 4;
    f32x2 a = *(const f32x2*)(&sH[w][ln * EMBED_DIM + k0 + koff]);   // A: M=ln, K
    f32x2 b = *(const f32x2*)(&sKW[ln * EMBED_DIM + k0 + koff]);     // B[c][q]=KW[q][c], q=ln
    dacc = __builtin_amdgcn_wmma_f32_16x16x4_f32(
        false, a, false, b, (short)0, dacc, false, false);
  }

  // Store logits: D element (v,l): clause m = v + 8*(l>>4), queue q = ln.
  const int q = ln;
  if (q < NUM_QUEUES) {
    const float kbq = kb[q];
#pragma unroll
    for (int v = 0; v < 8; ++v) {
      const int m = v + ((l >> 4) << 3);
      logits[(size_t)q * N_CLAUSES + r0 + m] = dacc[v] + kbq;
    }
  }
}

// ---------------------------------------------------------------------------
// Kernel 2: one block per selection step. Single-pass online masked softmax:
// maintain (m, S1=sum e^(x-m), S2=sum (x-m)e^(x-m), count); then
//   logZ_rel = log(S1); lsm_i = x_i - m - logZ_rel
//   ce = -(x_sel - m - logZ_rel)
//   minus_entropy = S2/S1 - logZ_rel;  normalized by log(count)
// Mask is a jax bool array -> 1 byte per element; read 4 at a time as uint.
// ---------------------------------------------------------------------------
#define LOSS_BLOCK 512

__global__ void plm_loss_kernel(const float* __restrict__ logits,
                                const unsigned char* __restrict__ mask,
                                const float* __restrict__ rewards,
                                const int* __restrict__ queue_idx,
                                const int* __restrict__ sel_idx,
                                float* __restrict__ partial) {
  const int s = blockIdx.x;
  const int q = queue_idx[s];
  const float* row = logits + (size_t)q * N_CLAUSES;
  const unsigned char* mrow = mask + (size_t)s * N_CLAUSES;

  float mloc = -INFINITY, s1 = 0.f, s2 = 0.f;
  int cnt = 0;

  // N_CLAUSES is a multiple of 4; 4 mask bytes per uint load, float4 logits.
  for (int j = threadIdx.x * 4; j < N_CLAUSES; j += LOSS_BLOCK * 4) {
    const unsigned int mb = *(const unsigned int*)(mrow + j);
    if (mb == 0u) continue;
    const f32x4 xs = *(const f32x4*)(row + j);
#pragma unroll
    for (int k = 0; k < 4; ++k) {
      if ((mb >> (8 * k)) & 0xffu) {
        const float x = xs[k];
        if (x > mloc) {
          if (s1 == 0.f) {
            mloc = x; s1 = 1.f;            // first member: s2 stays 0
          } else {
            const float d = mloc - x;      // < 0
            const float e = expf(d);
            s2 = e * (s2 + d * s1);
            s1 = e * s1 + 1.f;
            mloc = x;
          }
        } else {
          const float d = x - mloc;
          const float e = expf(d);
          s1 += e;
          s2 += e * d;
        }
        ++cnt;
      }
    }
  }

  __shared__ float sm[LOSS_BLOCK], ss1[LOSS_BLOCK], ss2[LOSS_BLOCK];
  __shared__ int sc[LOSS_BLOCK];
  const int tid = threadIdx.x;
  sm[tid] = mloc; ss1[tid] = s1; ss2[tid] = s2; sc[tid] = cnt;
  __syncthreads();

  for (int off = LOSS_BLOCK / 2; off > 0; off >>= 1) {
    if (tid < off) {
      const float ma = sm[tid],       s1a = ss1[tid],       s2a = ss2[tid];
      const float mb = sm[tid + off], s1b = ss1[tid + off], s2b = ss2[tid + off];
      float M, S1, S2;
      if (s1b == 0.f)      { M = ma; S1 = s1a; S2 = s2a; }
      else if (s1a == 0.f) { M = mb; S1 = s1b; S2 = s2b; }
      else {
        M = fmaxf(ma, mb);
        const float ea = expf(ma - M), eb = expf(mb - M);
        S1 = s1a * ea + s1b * eb;
        S2 = (s2a + (ma - M) * s1a) * ea + (s2b + (mb - M) * s1b) * eb;
      }
      sm[tid] = M; ss1[tid] = S1; ss2[tid] = S2;
      sc[tid] += sc[tid + off];
    }
    __syncthreads();
  }

  if (tid == 0) {
    const float M = sm[0], S1 = ss1[0], S2 = ss2[0];
    const float logS = logf(S1);
    const float xsel = row[sel_idx[s]];
    const float ce = -(xsel - M - logS);
    const float me = S2 / S1 - logS;                   // sum p*log p
    const float men = me / logf((float)sc[0]);         // ENTROPY_NORMALIZED
    partial[s] = rewards[s] * ce + ENTROPY_COEF * men;
  }
}

// ---------------------------------------------------------------------------
// Kernel 3: deterministic final reduction of the 2048 per-step partials.
// ---------------------------------------------------------------------------
__global__ void plm_reduce_kernel(const float* __restrict__ partial,
                                  float* __restrict__ out) {
  __shared__ float red[256];
  float acc = 0.f;
  for (int i = threadIdx.x; i < NUM_STEPS; i += 256) acc += partial[i];
  red[threadIdx.x] = acc;
  __syncthreads();
  for (int off = 128; off > 0; off >>= 1) {
    if (threadIdx.x < off) red[threadIdx.x] += red[threadIdx.x + off];
    __syncthreads();
  }
  if (threadIdx.x == 0) out[0] = red[0];
}

// ---------------------------------------------------------------------------
// Launch
// ---------------------------------------------------------------------------
extern "C" void kernel_launch(void* const* d_in, const int* in_sizes, int n_in,
                              void* d_out, int out_size, void* d_ws, size_t ws_size,
                              hipStream_t stream) {
  const float*         X        = (const float*)d_in[0];  // [50000,64]
  const float*         W1       = (const float*)d_in[1];  // [64,128]
  const float*         b1       = (const float*)d_in[2];  // [128]
  const float*         W2       = (const float*)d_in[3];  // [128,128]
  const float*         b2       = (const float*)d_in[4];  // [128]
  const float*         keys     = (const float*)d_in[5];  // [8,128]
  const float*         rewards  = (const float*)d_in[6];  // [2048]
  const unsigned char* mask     = (const unsigned char*)d_in[7];  // [2048,50000] bool
  const int*           queue_i  = (const int*)d_in[8];    // [2048]
  const int*           sel_i    = (const int*)d_in[9];    // [2048]

  float* ws      = (float*)d_ws;
  float* logits  = ws;                                  // 8*50000 = 400000 floats
  float* KW      = logits + (size_t)NUM_QUEUES * N_CLAUSES;  // 16*128
  float* kb      = KW + 16 * EMBED_DIM;                 // 16
  float* partial = kb + 16;                             // 2048

  plm_prep_kernel<<<16, 128, 0, stream>>>(keys, W2, b2, KW, kb);

  const int blocks = (N_TILES + WAVES_PER_BLOCK - 1) / WAVES_PER_BLOCK;  // 782
  plm_embed_logits_kernel<<<blocks, WAVES_PER_BLOCK * 32, 0, stream>>>(
      X, W1, b1, KW, kb, logits);

  plm_loss_kernel<<<NUM_STEPS, LOSS_BLOCK, 0, stream>>>(
      logits, mask, rewards, queue_i, sel_i, partial);

  plm_reduce_kernel<<<1, 256, 0, stream>>>(partial, (float*)d_out);
}